// FeatureDistanceLayer_80659485819604
// MI455X (gfx1250) — compile-verified
//
#include <hip/hip_runtime.h>
#include <hip/hip_bf16.h>

// Weighted L1 distance: dist[b,n] = sum_d softplus(w)_d * |q[b,d] - X[n,d]|
// B=64, N=50000, D=64, fp32.
//
// Key algebraic fold: softplus(w) > 0  =>  w*|q-x| == |w*q - w*x|.
// So we pre-scale q and X by w and the inner loop is 2 VALU ops/element
// (v_sub_f32 + v_add_f32 with |src| modifier) — the minimum for exact fp32.
//
// CDNA5 path: one TENSOR_LOAD_TO_LDS (TDM) per block DMAs a 256x64 fp32 tile
// of X into LDS (with D# padding: +4 DWORDs per 64 DWORDs => 272B row stride,
// bank-conflict-free ds_load_b128 row reads), synchronized with
// s_wait_tensorcnt 0. OOB rows at the grid edge read back as zeros per TDM
// semantics; those lanes simply don't store.

#define D_DIM     64
#define B_DIM     64
#define TILE_N    256
#define NTHREADS  256
#define XS_STRIDE 68   // 64 floats + 4-float pad (TDM pad_amount=4dw / pad_interval=64dw)

typedef __attribute__((ext_vector_type(4))) unsigned int v4u;
typedef __attribute__((ext_vector_type(4))) int          v4i;
typedef __attribute__((ext_vector_type(8))) int          v8i;

__device__ __forceinline__ float softplus_f(float x) {
  // log(1 + e^x), overflow-safe
  return (x > 20.0f) ? x : log1pf(__expf(x));
}

__global__ __launch_bounds__(NTHREADS)
void FeatureDistanceLayer_kernel(const float* __restrict__ q,
                                 const float* __restrict__ X,
                                 const float* __restrict__ fw,
                                 float* __restrict__ out,
                                 int N) {
  __shared__ __align__(16) float ws[D_DIM];
  __shared__ __align__(16) float qs[B_DIM * D_DIM];
  __shared__ __align__(16) float Xs[TILE_N * XS_STRIDE];

  const int tid  = threadIdx.x;
  const int row0 = blockIdx.x * TILE_N;

#if __has_builtin(__builtin_amdgcn_tensor_load_to_lds)
#define USED_TDM 1
  // Wave 0 issues one TDM descriptor for the whole X tile (EXEC-independent,
  // descriptor lives in SGPRs; all values here are wave-uniform).
  if (tid < 32) {
    unsigned long long gaddr =
        (unsigned long long)(const void*)X + (unsigned long long)row0 * (D_DIM * 4);
    unsigned lds_off = (unsigned)(unsigned long long)(void*)&Xs[0];
    unsigned rows    = (unsigned)(N - row0);  // tensor_dim1 from tile start (OOB -> 0)

    v4u g0;
    g0[0] = 1u;                                              // count=1, user mode
    g0[1] = lds_off;                                         // lds_addr (bytes)
    g0[2] = (unsigned)gaddr;                                 // global_addr[31:0]
    g0[3] = (unsigned)((gaddr >> 32) & 0x1FFFFFFull)         // global_addr[56:32]
            | (2u << 30);                                    // type=2 ("image")

    v8i g1;
    g1[0] = (int)((2u << 16)        // data_size = 2 -> 4 bytes
                | (1u << 20)        // pad_enable
                | (5u << 22)        // pad_interval = 5 -> every 64 DWORDs (one row)
                | (3u << 25));      // pad_amount = 3 -> 4 DWORDs (16 B)
    g1[1] = (int)(((unsigned)D_DIM & 0xFFFFu) << 16);        // tensor_dim0 lo16 (=64)
    g1[2] = (int)((rows & 0xFFFFu) << 16);                   // dim0 hi16=0 | dim1 lo16
    g1[3] = (int)((rows >> 16) | ((unsigned)D_DIM << 16));   // dim1 hi16 | tile_dim0=64
    g1[4] = (int)TILE_N;                                     // tile_dim1=256 | tile_dim2=0
    g1[5] = (int)D_DIM;                                      // tensor_dim0_stride = 64 elems
    g1[6] = 0;                                               // stride0 hi | stride1 lo
    g1[7] = 0;                                               // stride1 hi (unused, 2D)

    v4i g2 = {0, 0, 0, 0};
    v4i g3 = {0, 0, 0, 0};
#if defined(__clang_major__) && __clang_major__ >= 23
    v8i g4 = {0, 0, 0, 0, 0, 0, 0, 0};
    __builtin_amdgcn_tensor_load_to_lds(g0, g1, g2, g3, g4, 0);
#else
    __builtin_amdgcn_tensor_load_to_lds(g0, g1, g2, g3, 0);
#endif
  }
#endif

  // w = softplus(feature_weights)
  if (tid < D_DIM) ws[tid] = softplus_f(fw[tid]);
  __syncthreads();

  // q' = q * w  (16 KB tile shared by all 64x256 outputs of this block)
#pragma unroll
  for (int i = tid; i < B_DIM * D_DIM; i += NTHREADS) {
    qs[i] = q[i] * ws[i & (D_DIM - 1)];
  }

#if defined(USED_TDM)
#if __has_builtin(__builtin_amdgcn_s_wait_tensorcnt)
  __builtin_amdgcn_s_wait_tensorcnt(0);   // no-op for waves with TENSORcnt==0
#else
  asm volatile("s_wait_tensorcnt 0x0" ::: "memory");
#endif
#endif
  __syncthreads();

  const int  n     = row0 + tid;
  const bool valid = (n < N);

  // Pull this thread's X row into registers, scaled by w: x' = w * x.
  const float4* wp = reinterpret_cast<const float4*>(ws);
  float4 xr[16];
#if defined(USED_TDM)
  const float4* xp = reinterpret_cast<const float4*>(&Xs[tid * XS_STRIDE]);
#else
  const int     nc = valid ? n : (N - 1);
  const float4* xp = reinterpret_cast<const float4*>(X + (size_t)nc * D_DIM);
#endif
#pragma unroll
  for (int j = 0; j < 16; ++j) {
    float4 v = xp[j];
    float4 w4 = wp[j];
    v.x *= w4.x; v.y *= w4.y; v.z *= w4.z; v.w *= w4.w;
    xr[j] = v;
  }

  // For each of the 64 query rows: acc = sum_d |q'[b,d] - x'[d]|.
  // q' reads are wave-uniform ds_load_b128 broadcasts; 4 independent
  // accumulation chains hide VALU latency.
  const float4* qp = reinterpret_cast<const float4*>(qs);
#pragma unroll 2
  for (int b = 0; b < B_DIM; ++b) {
    float a0 = 0.f, a1 = 0.f, a2 = 0.f, a3 = 0.f;
#pragma unroll
    for (int j = 0; j < 16; ++j) {
      float4 qv = qp[b * 16 + j];
      a0 += __builtin_fabsf(qv.x - xr[j].x);
      a1 += __builtin_fabsf(qv.y - xr[j].y);
      a2 += __builtin_fabsf(qv.z - xr[j].z);
      a3 += __builtin_fabsf(qv.w - xr[j].w);
    }
    if (valid) out[(size_t)b * N + n] = (a0 + a1) + (a2 + a3);
  }
}

extern "C" void kernel_launch(void* const* d_in, const int* in_sizes, int n_in,
                              void* d_out, int out_size, void* d_ws, size_t ws_size,
                              hipStream_t stream) {
  (void)n_in; (void)out_size; (void)d_ws; (void)ws_size;
  const float* q  = (const float*)d_in[0];   // [64, 64]
  const float* X  = (const float*)d_in[1];   // [N, 64]
  const float* fw = (const float*)d_in[2];   // [64]
  float* out = (float*)d_out;                // [64, N]

  const int N    = in_sizes[1] / D_DIM;
  const int grid = (N + TILE_N - 1) / TILE_N;

  hipLaunchKernelGGL(FeatureDistanceLayer_kernel,
                     dim3(grid), dim3(NTHREADS), 0, stream,
                     q, X, fw, out, N);
}